// Attn_31207232373419
// MI455X (gfx1250) — compile-verified
//
#include <hip/hip_runtime.h>
#include <hip/hip_bf16.h>
#include <math.h>

#define BB 32
#define LL 2048
#define HH 1024

typedef __attribute__((ext_vector_type(2))) float v2f;
typedef __attribute__((ext_vector_type(8))) float v8f;

// ---------------------------------------------------------------------------
// Kernel 1: u[h] = sum_d W_v[d] * W_attn[d, H + h]
// 1024 threads; for each d, threads read contiguous h -> fully coalesced.
// ---------------------------------------------------------------------------
__global__ __launch_bounds__(256) void attn_u_kernel(
    const float* __restrict__ Wattn,   // (H, 2H) row-major
    const float* __restrict__ Wv,      // (H)
    float* __restrict__ u)             // (H)
{
    int h = blockIdx.x * 256 + threadIdx.x;
    float acc = 0.0f;
    const float* col = Wattn + (size_t)HH + h;   // column h of Wa_e
    for (int d = 0; d < HH; ++d) {
        acc = fmaf(Wv[d], col[(size_t)d * (2 * HH)], acc);
    }
    u[h] = acc;
}

// ---------------------------------------------------------------------------
// Kernel 2: scores[b,l] = enc[l,b,:] . u   via V_WMMA_F32_16X16X4_F32
// Row r = l*B + b (contiguous H floats). Each wave owns 16 consecutive rows.
// A tile (16x4 f32): lanes 0-15 hold {K0,K1} of row M=lane, lanes 16-31 hold
// {K2,K3} of row M=lane-16.  B tile (4x16): only column 0 nonzero = u chunk.
// D column 0 -> lane 0 (rows 0-7, VGPR j) and lane 16 (rows 8-15, VGPR j).
// ---------------------------------------------------------------------------
__global__ __launch_bounds__(256) void attn_scores_wmma_kernel(
    const float* __restrict__ enc,     // (L, B, H)
    const float* __restrict__ u,       // (H)
    float* __restrict__ scores)        // (B, L) row-major
{
    __shared__ float s_u[HH];
    int tid = threadIdx.x;
    #pragma unroll
    for (int i = 0; i < HH / 256; ++i) s_u[tid + i * 256] = u[tid + i * 256];
    __syncthreads();

    const int wave = tid >> 5;          // 0..7
    const int lane = tid & 31;
    const int half = lane >> 4;         // 0 or 1 (selects K pair)
    const int lm   = lane & 15;         // row within 16-row group
    const long r0  = ((long)blockIdx.x * 8 + wave) * 16;

    const float* aBase = enc + (r0 + lm) * (long)HH + half * 2;
    const bool  bLane  = (lm == 0);     // only lanes 0 and 16 carry B data

    v8f acc = {0.f, 0.f, 0.f, 0.f, 0.f, 0.f, 0.f, 0.f};

    for (int kk = 0; kk < HH; kk += 16) {
        #pragma unroll
        for (int s = 0; s < 4; ++s) {
            const int k = kk + s * 4;
            v2f a = *(const v2f*)(aBase + k);                 // b64 load
            v2f bu = *(const v2f*)(&s_u[k + half * 2]);       // ds_load_b64
            v2f b;
            b.x = bLane ? bu.x : 0.0f;
            b.y = bLane ? bu.y : 0.0f;
            acc = __builtin_amdgcn_wmma_f32_16x16x4_f32(
                /*neg_a=*/false, a, /*neg_b=*/false, b,
                /*c_mod=*/(short)0, acc, /*reuse_a=*/false, /*reuse_b=*/false);
        }
    }

    // Column 0 of D lives in lanes 0 (M=0..7) and 16 (M=8..15), VGPRs 0..7.
    if (lm == 0) {
        #pragma unroll
        for (int j = 0; j < 8; ++j) {
            long r  = r0 + (long)half * 8 + j;   // global row = l*B + b
            int  b  = (int)(r & (BB - 1));
            long l  = r >> 5;                    // r / B
            scores[(long)b * LL + l] = acc[j];
        }
    }
}

// ---------------------------------------------------------------------------
// Kernel 3: in-place row softmax over l. One block (256 thr) per batch row.
// Each thread holds its 8 values in registers before any write -> in-place OK.
// ---------------------------------------------------------------------------
__global__ __launch_bounds__(256) void attn_softmax_kernel(float* __restrict__ data)
{
    __shared__ float red[16];
    const int tid = threadIdx.x;
    float* row = data + (long)blockIdx.x * LL;

    float vals[8];
    float m = -INFINITY;
    #pragma unroll
    for (int i = 0; i < 8; ++i) {
        vals[i] = row[tid + i * 256];
        m = fmaxf(m, vals[i]);
    }
    #pragma unroll
    for (int off = 16; off > 0; off >>= 1)
        m = fmaxf(m, __shfl_xor(m, off, 32));
    if ((tid & 31) == 0) red[tid >> 5] = m;
    __syncthreads();
    m = red[0];
    #pragma unroll
    for (int w = 1; w < 8; ++w) m = fmaxf(m, red[w]);

    float s = 0.0f;
    #pragma unroll
    for (int i = 0; i < 8; ++i) {
        vals[i] = __expf(vals[i] - m);
        s += vals[i];
    }
    #pragma unroll
    for (int off = 16; off > 0; off >>= 1)
        s += __shfl_xor(s, off, 32);
    if ((tid & 31) == 0) red[8 + (tid >> 5)] = s;
    __syncthreads();
    float tot = 0.0f;
    #pragma unroll
    for (int w = 0; w < 8; ++w) tot += red[8 + w];
    const float inv = 1.0f / tot;

    #pragma unroll
    for (int i = 0; i < 8; ++i)
        row[tid + i * 256] = vals[i] * inv;
}

// ---------------------------------------------------------------------------
extern "C" void kernel_launch(void* const* d_in, const int* in_sizes, int n_in,
                              void* d_out, int out_size, void* d_ws, size_t ws_size,
                              hipStream_t stream)
{
    // setup_inputs order: hidden(0), encoder_outputs(1), W_attn(2), b_attn(3),
    // W_v(4), b_v(5). hidden/b_attn/b_v contribute only a per-row constant to
    // the scores, which softmax is exactly invariant to -> not needed.
    const float* enc   = (const float*)d_in[1];
    const float* Wattn = (const float*)d_in[2];
    const float* Wv    = (const float*)d_in[4];
    float* out = (float*)d_out;            // (B, L) = 65536 floats

    float* u = (float*)d_ws;               // H floats of scratch

    attn_u_kernel<<<HH / 256, 256, 0, stream>>>(Wattn, Wv, u);

    // (L*B)/16 rows-per-wave, 8 waves per block -> 512 blocks
    const int nRows   = LL * BB;
    const int nBlocks = nRows / (16 * 8);
    attn_scores_wmma_kernel<<<nBlocks, 256, 0, stream>>>(enc, u, out);

    attn_softmax_kernel<<<BB, 256, 0, stream>>>(out);
}